// Decoder_86663850098731
// MI455X (gfx1250) — compile-verified
//
#include <hip/hip_runtime.h>
#include <math.h>

typedef __attribute__((ext_vector_type(2))) float v2f;
typedef __attribute__((ext_vector_type(8))) float v8f;

#define B_  32
#define NX  2048
#define NC  1024
#define NM  2560   // 512 hole + 2048 x

__device__ __forceinline__ v8f wmma_f32_step(v2f a, v2f b, v8f c) {
  // D = A(16x4, f32) x B(4x16, f32) + C(16x16, f32)
  return __builtin_amdgcn_wmma_f32_16x16x4_f32(false, a, false, b, (short)0, c,
                                               false, false);
}

// ---------------------------------------------------------------------------
// GEMM: out[M,N] = act(A[M,K] @ Wt[N,K]^T + bias[N]); one wave per 16x16 tile.
// grid = (M/16, N/16), block = 32 (one wave32).
// ---------------------------------------------------------------------------
__global__ void k_gemm_rowbias(const float* __restrict__ A,
                               const float* __restrict__ Wt,
                               const float* __restrict__ bias,
                               float* __restrict__ out, int K, int N, int relu) {
  int lane = threadIdx.x;
  int half = lane >> 4, lr = lane & 15;
  int m0 = blockIdx.x * 16, n0 = blockIdx.y * 16;
  v8f c = {};
  for (int k = 0; k < K; k += 4) {
    int ka = k + 2 * half;
    v2f a, b;
    a.x = A[(m0 + lr) * K + ka];
    a.y = A[(m0 + lr) * K + ka + 1];
    b.x = Wt[(n0 + lr) * K + ka];
    b.y = Wt[(n0 + lr) * K + ka + 1];
    c = wmma_f32_step(a, b, c);
  }
#pragma unroll
  for (int v = 0; v < 8; v++) {
    int row = m0 + v + 8 * half;
    int col = n0 + lr;
    float val = c[v] + bias[col];
    if (relu) val = fmaxf(val, 0.f);
    out[row * N + col] = val;
  }
}

// coarse [B,3072] (== [B,3,1024]) -> d_out first part [B,1024,3]
__global__ void k_coarse_out(const float* __restrict__ coarse,
                             float* __restrict__ outp) {
  int i = blockIdx.x * blockDim.x + threadIdx.x;
  if (i >= B_ * NC * 3) return;
  int b = i / (NC * 3);
  int rem = i % (NC * 3);
  int n = rem / 3, c = rem % 3;
  outp[i] = coarse[b * 3072 + c * NC + n];
}

// dist1[b,i] = min_j ( |c_i|^2 + |x_j|^2 - 2 c_i.x_j )   (sqdist formula)
__global__ void k_chamfer(const float* __restrict__ coarse,
                          const float* __restrict__ x,
                          float* __restrict__ dist1) {
  int gid = blockIdx.x * blockDim.x + threadIdx.x;
  if (gid >= B_ * NC) return;
  int b = gid / NC, i = gid % NC;
  const float* C = coarse + b * 3072;
  float cx = C[i], cy = C[NC + i], cz = C[2 * NC + i];
  float cc = cx * cx + cy * cy + cz * cz;
  const float* X = x + b * 3 * NX;
  float best = __builtin_inff();
  for (int j = 0; j < NX; j++) {
    float xx_ = X[j], xy = X[NX + j], xz = X[2 * NX + j];
    float d = cc + (xx_ * xx_ + xy * xy + xz * xz)
                 - 2.f * (cx * xx_ + cy * xy + cz * xz);
    best = fminf(best, d);
  }
  dist1[gid] = best;
}

// per-batch top-512 largest, sorted descending, stable (lowest index on ties)
__global__ __launch_bounds__(256) void k_top512(const float* __restrict__ dist1,
                                                int* __restrict__ ihole) {
  __shared__ float v[NC];
  __shared__ float rv[256];
  __shared__ int ri[256];
  int b = blockIdx.x, t = threadIdx.x;
  for (int i = t; i < NC; i += 256) v[i] = dist1[b * NC + i];
  __syncthreads();
  for (int s = 0; s < 512; s++) {
    float bv = -__builtin_inff();
    int bi = 0x7fffffff;
    for (int i = t; i < NC; i += 256) {
      float x = v[i];
      if (x > bv) { bv = x; bi = i; }  // i ascending -> ties keep first
    }
    rv[t] = bv; ri[t] = bi;
    __syncthreads();
    for (int off = 128; off > 0; off >>= 1) {
      if (t < off) {
        if (rv[t + off] > rv[t] ||
            (rv[t + off] == rv[t] && ri[t + off] < ri[t])) {
          rv[t] = rv[t + off]; ri[t] = ri[t + off];
        }
      }
      __syncthreads();
    }
    if (t == 0) { ihole[b * 512 + s] = ri[0]; v[ri[0]] = -__builtin_inff(); }
    __syncthreads();
  }
}

// merge[b,d,:] = [ coarse[b,d,ihole], x[b,d,:] ]
__global__ void k_merge(const float* __restrict__ coarse,
                        const float* __restrict__ x,
                        const int* __restrict__ ihole,
                        float* __restrict__ merge) {
  int gid = blockIdx.x * blockDim.x + threadIdx.x;
  if (gid >= B_ * 3 * NM) return;
  int b = gid / (3 * NM);
  int rem = gid % (3 * NM);
  int d = rem / NM, j = rem % NM;
  float val;
  if (j < 512)
    val = coarse[b * 3072 + d * NC + ihole[b * 512 + j]];
  else
    val = x[b * 3 * NX + d * NX + (j - 512)];
  merge[gid] = val;
}

// farthest point sampling: one workgroup per batch, serial 1024 rounds
__global__ __launch_bounds__(256) void k_fps(const float* __restrict__ merge,
                                             int* __restrict__ fidx) {
  __shared__ float px[NM], py[NM], pz[NM], dd[NM];
  __shared__ float rv[256];
  __shared__ int ri[256];
  __shared__ int last_s;
  int b = blockIdx.x, t = threadIdx.x;
  const float* M = merge + b * 3 * NM;
  for (int i = t; i < NM; i += 256) {
    px[i] = M[i]; py[i] = M[NM + i]; pz[i] = M[2 * NM + i];
    dd[i] = __builtin_inff();
  }
  if (t == 0) last_s = 0;
  __syncthreads();
  for (int s = 0; s < NC; s++) {
    int last = last_s;
    if (t == 0) fidx[b * NC + s] = last;
    float sx = px[last], sy = py[last], sz = pz[last];
    float bv = -__builtin_inff();
    int bi = 0x7fffffff;
    for (int i = t; i < NM; i += 256) {
      float dx = px[i] - sx, dy = py[i] - sy, dz = pz[i] - sz;
      float nd = fminf(dd[i], dx * dx + dy * dy + dz * dz);
      dd[i] = nd;
      if (nd > bv) { bv = nd; bi = i; }  // first occurrence on ties
    }
    rv[t] = bv; ri[t] = bi;
    __syncthreads();
    for (int off = 128; off > 0; off >>= 1) {
      if (t < off) {
        if (rv[t + off] > rv[t] ||
            (rv[t + off] == rv[t] && ri[t + off] < ri[t])) {
          rv[t] = rv[t + off]; ri[t] = ri[t + off];
        }
      }
      __syncthreads();
    }
    if (t == 0) last_s = ri[0];
    __syncthreads();
  }
}

__global__ void k_gather_xx(const float* __restrict__ merge,
                            const int* __restrict__ fidx,
                            float* __restrict__ xx) {
  int gid = blockIdx.x * blockDim.x + threadIdx.x;
  if (gid >= B_ * 3 * NC) return;
  int b = gid / (3 * NC);
  int rem = gid % (3 * NC);
  int d = rem / NC, n = rem % NC;
  xx[gid] = merge[b * 3 * NM + d * NM + fidx[b * NC + n]];
}

// kNN(8, self included) + 3x3 covariance -> covm [B,9,N], idx [B,N,8]
__global__ void k_knn_cov(const float* __restrict__ xx,
                          int* __restrict__ knnidx, float* __restrict__ covm) {
  int gid = blockIdx.x * blockDim.x + threadIdx.x;
  if (gid >= B_ * NC) return;
  int b = gid / NC, i = gid % NC;
  const float* X = xx + b * 3 * NC;
  float pix = X[i], piy = X[NC + i], piz = X[2 * NC + i];
  float ni = pix * pix + piy * piy + piz * piz;
  float kd[8]; int ki[8];
#pragma unroll
  for (int q = 0; q < 8; q++) { kd[q] = __builtin_inff(); ki[q] = 0; }
  for (int j = 0; j < NC; j++) {
    float qx = X[j], qy = X[NC + j], qz = X[2 * NC + j];
    float nj = qx * qx + qy * qy + qz * qz;
    float d = ni + nj - 2.f * (pix * qx + piy * qy + piz * qz);
    if (d < kd[7]) {                    // strict: equal-dist later idx dropped
      kd[7] = d; ki[7] = j;
#pragma unroll
      for (int q = 7; q > 0; q--) {     // bubble new element to place (stable)
        if (kd[q] < kd[q - 1]) {
          float td = kd[q]; kd[q] = kd[q - 1]; kd[q - 1] = td;
          int ti = ki[q]; ki[q] = ki[q - 1]; ki[q - 1] = ti;
        }
      }
    }
  }
  float nx[8], ny[8], nz[8];
  float mx = 0.f, my = 0.f, mz = 0.f;
#pragma unroll
  for (int q = 0; q < 8; q++) {
    int jj = ki[q];
    nx[q] = X[jj]; ny[q] = X[NC + jj]; nz[q] = X[2 * NC + jj];
    mx += nx[q]; my += ny[q]; mz += nz[q];
    knnidx[(b * NC + i) * 8 + q] = jj;
  }
  mx *= 0.125f; my *= 0.125f; mz *= 0.125f;
  float cxx = 0, cxy = 0, cxz = 0, cyy = 0, cyz = 0, czz = 0;
#pragma unroll
  for (int q = 0; q < 8; q++) {
    float dx = nx[q] - mx, dy = ny[q] - my, dz = nz[q] - mz;
    cxx += dx * dx; cxy += dx * dy; cxz += dx * dz;
    cyy += dy * dy; cyz += dy * dz; czz += dz * dz;
  }
  float* CV = covm + b * 9 * NC;
  CV[0 * NC + i] = cxx * 0.125f; CV[1 * NC + i] = cxy * 0.125f;
  CV[2 * NC + i] = cxz * 0.125f; CV[3 * NC + i] = cxy * 0.125f;
  CV[4 * NC + i] = cyy * 0.125f; CV[5 * NC + i] = cyz * 0.125f;
  CV[6 * NC + i] = cxz * 0.125f; CV[7 * NC + i] = cyz * 0.125f;
  CV[8 * NC + i] = czz * 0.125f;
}

// m1: 3 -> 32 (relu) -> 64 (relu), per point
__global__ void k_m1(const float* __restrict__ xx, const float* __restrict__ w1,
                     const float* __restrict__ b1, const float* __restrict__ w2,
                     const float* __restrict__ b2, float* __restrict__ f1) {
  int gid = blockIdx.x * blockDim.x + threadIdx.x;
  if (gid >= B_ * NC) return;
  int b = gid / NC, n = gid % NC;
  const float* X = xx + b * 3 * NC;
  float x0 = X[n], x1 = X[NC + n], x2 = X[2 * NC + n];
  float h[32];
#pragma unroll
  for (int o = 0; o < 32; o++)
    h[o] = fmaxf(w1[o * 3] * x0 + w1[o * 3 + 1] * x1 + w1[o * 3 + 2] * x2 +
                 b1[o], 0.f);
  float* F = f1 + b * 64 * NC;
  for (int o = 0; o < 64; o++) {
    float acc = b2[o];
#pragma unroll
    for (int c = 0; c < 32; c++) acc += w2[o * 32 + c] * h[c];
    F[o * NC + n] = fmaxf(acc, 0.f);
  }
}

// ---------------------------------------------------------------------------
// Fused edge-attention: per workgroup = (batch, 16 points).
// E[112][64]   : edge features (7 nbrs x 16 pts rows), staged in LDS
// WMMA GEMM1   : H = relu(E @ att_w1^T + b1)     (M=112,K=64,N=128)
// WMMA GEMM2   : W = H @ att_w2^T + b2           (M=112,K=128,N=64)
// softmax over 7 neighbors per (point, channel), weighted sum of E.
// ---------------------------------------------------------------------------
__global__ __launch_bounds__(256) void k_attention(
    const float* __restrict__ f1, const int* __restrict__ knnidx,
    const float* __restrict__ w1, const float* __restrict__ b1,
    const float* __restrict__ w2, const float* __restrict__ b2,
    float* __restrict__ f1out) {
  __shared__ float E[112][64];
  __shared__ float H[112][128];
  __shared__ float WT[112][64];
  int t = threadIdx.x;
  int b = blockIdx.y, n0 = blockIdx.x * 16;
  const float* F = f1 + b * 64 * NC;
  // phase 1: stage edge vectors
  for (int e = t; e < 112 * 64; e += 256) {
    int m = e >> 6, c = e & 63;
    int j = m >> 4, p = m & 15;   // row m = j*16 + p
    int n = n0 + p;
    int nb = knnidx[(b * NC + n) * 8 + (j + 1)];
    E[m][c] = F[c * NC + nb] - F[c * NC + n];
  }
  __syncthreads();
  int wave = t >> 5, lane = t & 31, half = lane >> 4, lr = lane & 15;
  // phase 2: GEMM1, 56 tiles over 8 waves
  for (int tt = wave; tt < 7 * 8; tt += 8) {
    int Mt = tt >> 3, Nt = tt & 7;
    v8f c = {};
    for (int k = 0; k < 64; k += 4) {
      int ka = k + 2 * half;
      v2f a, bf;
      a.x = E[Mt * 16 + lr][ka];
      a.y = E[Mt * 16 + lr][ka + 1];
      bf.x = w1[(Nt * 16 + lr) * 64 + ka];
      bf.y = w1[(Nt * 16 + lr) * 64 + ka + 1];
      c = wmma_f32_step(a, bf, c);
    }
#pragma unroll
    for (int v = 0; v < 8; v++)
      H[Mt * 16 + v + 8 * half][Nt * 16 + lr] =
          fmaxf(c[v] + b1[Nt * 16 + lr], 0.f);
  }
  __syncthreads();
  // phase 3: GEMM2, 28 tiles over 8 waves
  for (int tt = wave; tt < 7 * 4; tt += 8) {
    int Mt = tt >> 2, Nt = tt & 3;
    v8f c = {};
    for (int k = 0; k < 128; k += 4) {
      int ka = k + 2 * half;
      v2f a, bf;
      a.x = H[Mt * 16 + lr][ka];
      a.y = H[Mt * 16 + lr][ka + 1];
      bf.x = w2[(Nt * 16 + lr) * 128 + ka];
      bf.y = w2[(Nt * 16 + lr) * 128 + ka + 1];
      c = wmma_f32_step(a, bf, c);
    }
#pragma unroll
    for (int v = 0; v < 8; v++)
      WT[Mt * 16 + v + 8 * half][Nt * 16 + lr] = c[v] + b2[Nt * 16 + lr];
  }
  __syncthreads();
  // phase 4: softmax over neighbors + weighted sum
  float* FO = f1out + b * 64 * NC;
  for (int e = t; e < 16 * 64; e += 256) {
    int p = e >> 6, c = e & 63;
    float mx = -__builtin_inff();
#pragma unroll
    for (int j = 0; j < 7; j++) mx = fmaxf(mx, WT[j * 16 + p][c]);
    float se = 0.f, acc = 0.f;
#pragma unroll
    for (int j = 0; j < 7; j++) {
      float w = expf(WT[j * 16 + p][c] - mx);
      se += w;
      acc += w * E[j * 16 + p][c];
    }
    FO[c * NC + (n0 + p)] = acc / se;
  }
}

// fcat[0:64] = attention output; fcat[64:96] = relu(m2_w1 @ covm + b)
__global__ void k_m2cat(const float* __restrict__ f1b,
                        const float* __restrict__ covm,
                        const float* __restrict__ w,
                        const float* __restrict__ bias,
                        float* __restrict__ fcat) {
  int gid = blockIdx.x * blockDim.x + threadIdx.x;
  if (gid >= B_ * NC) return;
  int b = gid / NC, n = gid % NC;
  const float* FI = f1b + b * 64 * NC;
  float* FO = fcat + b * 96 * NC;
  for (int c = 0; c < 64; c++) FO[c * NC + n] = FI[c * NC + n];
  const float* CV = covm + b * 9 * NC;
  float cv[9];
#pragma unroll
  for (int e = 0; e < 9; e++) cv[e] = CV[e * NC + n];
  for (int o = 0; o < 32; o++) {
    float acc = bias[o];
#pragma unroll
    for (int e = 0; e < 9; e++) acc += w[o * 9 + e] * cv[e];
    FO[(64 + o) * NC + n] = fmaxf(acc, 0.f);
  }
}

// m3 layer1 via WMMA: hid[b,128,N] = relu(W[128,96] @ fcat[b,96,N] + b)
// grid = (8 Mtiles, 64 Ntiles, B), block = 32. Bias is on the M (row) dim.
__global__ void k_m3a(const float* __restrict__ W, const float* __restrict__ bias,
                      const float* __restrict__ fcat, float* __restrict__ hid) {
  int lane = threadIdx.x, half = lane >> 4, lr = lane & 15;
  int m0 = blockIdx.x * 16, n0 = blockIdx.y * 16, b = blockIdx.z;
  const float* F = fcat + b * 96 * NC;
  v8f c = {};
  for (int k = 0; k < 96; k += 4) {
    int ka = k + 2 * half;
    v2f a, bf;
    a.x = W[(m0 + lr) * 96 + ka];
    a.y = W[(m0 + lr) * 96 + ka + 1];
    bf.x = F[ka * NC + n0 + lr];
    bf.y = F[(ka + 1) * NC + n0 + lr];
    c = wmma_f32_step(a, bf, c);
  }
  float* Hb = hid + b * 128 * NC;
#pragma unroll
  for (int v = 0; v < 8; v++) {
    int row = m0 + v + 8 * half;
    Hb[row * NC + n0 + lr] = fmaxf(c[v] + bias[row], 0.f);
  }
}

// m3 layer2 (128->6) + reshape(B,3,2048) + tile(xx)+0.15*delta + transpose
__global__ void k_final(const float* __restrict__ hid,
                        const float* __restrict__ xx,
                        const float* __restrict__ w, const float* __restrict__ b6,
                        float* __restrict__ outp) {
  int gid = blockIdx.x * blockDim.x + threadIdx.x;
  if (gid >= B_ * 2 * NC) return;
  int b = gid / (2 * NC), tt = gid % (2 * NC);
  int n = tt & (NC - 1);
  int hi = tt >> 10;  // 0 or 1
  const float* Hb = hid + b * 128 * NC;
  float* o = outp + (size_t)(b * 2 * NC + tt) * 3;
#pragma unroll
  for (int d = 0; d < 3; d++) {
    int c6 = 2 * d + hi;  // delta.reshape(B,3,2048): flat d*2048+tt == c6*1024+n
    float acc = b6[c6];
    for (int c = 0; c < 128; c++) acc += w[c6 * 128 + c] * Hb[c * NC + n];
    o[d] = xx[b * 3 * NC + d * NC + n] + 0.15f * acc;
  }
}

extern "C" void kernel_launch(void* const* d_in, const int* in_sizes, int n_in,
                              void* d_out, int out_size, void* d_ws,
                              size_t ws_size, hipStream_t stream) {
  (void)in_sizes; (void)n_in; (void)out_size; (void)ws_size;
  const float* g      = (const float*)d_in[0];
  const float* x      = (const float*)d_in[1];
  const float* lin_w1 = (const float*)d_in[2];
  const float* lin_b1 = (const float*)d_in[3];
  const float* lin_w2 = (const float*)d_in[4];
  const float* lin_b2 = (const float*)d_in[5];
  const float* m1_w1  = (const float*)d_in[6];
  const float* m1_b1  = (const float*)d_in[7];
  const float* m1_w2  = (const float*)d_in[8];
  const float* m1_b2  = (const float*)d_in[9];
  const float* m2_w1  = (const float*)d_in[10];
  const float* m2_b1  = (const float*)d_in[11];
  const float* m3_w1  = (const float*)d_in[12];
  const float* m3_b1  = (const float*)d_in[13];
  const float* m3_w2  = (const float*)d_in[14];
  const float* m3_b2  = (const float*)d_in[15];
  const float* att_w1 = (const float*)d_in[16];
  const float* att_b1 = (const float*)d_in[17];
  const float* att_w2 = (const float*)d_in[18];
  const float* att_b2 = (const float*)d_in[19];
  float* outp = (float*)d_out;
  char* ws = (char*)d_ws;

  float* h      = (float*)(ws + 0);         // 32*1024            f32
  float* coarse = (float*)(ws + 131072);    // 32*3072
  float* dist1  = (float*)(ws + 524288);    // 32*1024
  int*   ihole  = (int*)  (ws + 655360);    // 32*512
  float* merge  = (float*)(ws + 720896);    // 32*3*2560
  int*   fidx   = (int*)  (ws + 1703936);   // 32*1024
  float* xx     = (float*)(ws + 1835008);   // 32*3*1024
  int*   knnidx = (int*)  (ws + 2228224);   // 32*1024*8
  float* covm   = (float*)(ws + 3276800);   // 32*9*1024
  float* f1     = (float*)(ws + 4456448);   // 32*64*1024
  float* f1b    = (float*)(ws + 12845056);  // 32*64*1024
  float* fcat   = (float*)(ws + 21233664);  // 32*96*1024
  float* hid    = (float*)(ws + 33816576);  // 32*128*1024  (end ~50.6 MB)

  k_gemm_rowbias<<<dim3(2, 64), 32, 0, stream>>>(g, lin_w1, lin_b1, h, 1024,
                                                 1024, 1);
  k_gemm_rowbias<<<dim3(2, 192), 32, 0, stream>>>(h, lin_w2, lin_b2, coarse,
                                                  1024, 3072, 0);
  k_coarse_out<<<384, 256, 0, stream>>>(coarse, outp);
  k_chamfer<<<128, 256, 0, stream>>>(coarse, x, dist1);
  k_top512<<<32, 256, 0, stream>>>(dist1, ihole);
  k_merge<<<960, 256, 0, stream>>>(coarse, x, ihole, merge);
  k_fps<<<32, 256, 0, stream>>>(merge, fidx);
  k_gather_xx<<<384, 256, 0, stream>>>(merge, fidx, xx);
  k_knn_cov<<<128, 256, 0, stream>>>(xx, knnidx, covm);
  k_m1<<<128, 256, 0, stream>>>(xx, m1_w1, m1_b1, m1_w2, m1_b2, f1);
  k_attention<<<dim3(64, 32), 256, 0, stream>>>(f1, knnidx, att_w1, att_b1,
                                                att_w2, att_b2, f1b);
  k_m2cat<<<128, 256, 0, stream>>>(f1b, covm, m2_w1, m2_b1, fcat);
  k_m3a<<<dim3(8, 64, 32), 32, 0, stream>>>(m3_w1, m3_b1, fcat, hid);
  k_final<<<256, 256, 0, stream>>>(hid, xx, m3_w2, m3_b2, outp + B_ * NC * 3);
}